// opt_policy_56831007261150
// MI455X (gfx1250) — compile-verified
//
#include <hip/hip_runtime.h>
#include <hip/hip_bf16.h>
#include <math.h>

typedef float v2f __attribute__((ext_vector_type(2)));
typedef float v8f __attribute__((ext_vector_type(8)));

#define HDIM 32

// ---------- helpers ----------
__device__ __forceinline__ unsigned fflip(float f) {
    unsigned u = __float_as_uint(f);
    return (u >> 31) ? ~u : (u | 0x80000000u);   // order-preserving map float->uint
}
__device__ __forceinline__ float funflip(unsigned u) {
    return __uint_as_float((u >> 31) ? (u & 0x7FFFFFFFu) : ~u);
}

// ---------- init: deg=1 (self loop), reduction scalars ----------
__global__ void k_init(unsigned* __restrict__ deg, unsigned* __restrict__ maxbits,
                       float* __restrict__ sumf, int N) {
    int i = blockIdx.x * blockDim.x + threadIdx.x;
    if (i < N) deg[i] = 1u;
    if (i == 0) { *maxbits = fflip(-INFINITY); *sumf = 0.0f; }
}

// ---------- degree count over edges ----------
__global__ void k_deg(const int* __restrict__ dst, unsigned* __restrict__ deg, int E) {
    int e = blockIdx.x * blockDim.x + threadIdx.x;
    if (e < E) atomicAdd(&deg[dst[e]], 1u);
}

__global__ void k_dinv(const unsigned* __restrict__ deg, float* __restrict__ dinv, int N) {
    int i = blockIdx.x * blockDim.x + threadIdx.x;
    if (i < N) dinv[i] = rsqrtf((float)deg[i]);
}

// ---------- WMMA f32 GEMM:  Y[N x 32] = X[N x K] @ W[K x 32] ----------
// One wave per 16-row tile; two 16x16 accumulators cover the 32 output cols.
// A layout (16x4 f32): lanes 0-15 hold M=lane, K={k0,k0+1}; lanes 16-31 hold M=lane-16, K={k0+2,k0+3}
// B layout (4x16 f32): lanes 0-15 hold N=lane, K={k0,k0+1}; lanes 16-31 hold N=lane-16, K={k0+2,k0+3}
__global__ void k_gemm_wmma(const float* __restrict__ X, const float* __restrict__ W,
                            float* __restrict__ Y, int N, int K) {
    int wave = (blockIdx.x * blockDim.x + threadIdx.x) >> 5;
    int lane = threadIdx.x & 31;
    int row0 = wave * 16;
    if (row0 >= N) return;

    int half = lane >> 4;            // 0: lanes 0-15, 1: lanes 16-31
    int l16  = lane & 15;
    int arow = row0 + l16; if (arow >= N) arow = N - 1;   // clamp (EXEC must stay all-1 for WMMA)
    const float* aptr = X + (size_t)arow * K + half * 2;

    v8f c0 = {}; v8f c1 = {};
    for (int k0 = 0; k0 < K; k0 += 4) {
        v2f a = *(const v2f*)(aptr + k0);
        int krow = k0 + half * 2;
        v2f b0, b1;
        b0.x = W[(size_t)krow * HDIM + l16];
        b0.y = W[(size_t)(krow + 1) * HDIM + l16];
        b1.x = W[(size_t)krow * HDIM + 16 + l16];
        b1.y = W[(size_t)(krow + 1) * HDIM + 16 + l16];
        c0 = __builtin_amdgcn_wmma_f32_16x16x4_f32(false, a, false, b0, (short)0, c0, false, false);
        c1 = __builtin_amdgcn_wmma_f32_16x16x4_f32(false, a, false, b1, (short)0, c1, false, false);
    }
    // C/D layout: VGPR r, lanes 0-15 -> (M=r, N=l16); lanes 16-31 -> (M=r+8, N=l16)
    #pragma unroll
    for (int r = 0; r < 8; ++r) {
        int row = row0 + r + 8 * half;
        if (row < N) {
            Y[(size_t)row * HDIM + l16]      = c0[r];
            Y[(size_t)row * HDIM + 16 + l16] = c1[r];
        }
    }
}

// ---------- agg init: self-loop contribution + bias ----------
__global__ void k_agg_init(const float* __restrict__ hW, const float* __restrict__ dinv,
                           const float* __restrict__ bias, float* __restrict__ agg, int N) {
    int i = blockIdx.x * blockDim.x + threadIdx.x;     // over N*32
    if (i >= N * HDIM) return;
    int node = i >> 5, c = i & (HDIM - 1);
    float di = dinv[node];
    agg[i] = hW[i] * di * di + bias[c];
}

// ---------- edge scatter: one wave per edge, one channel per lane ----------
__global__ void k_edge_agg(const float* __restrict__ hW, const int* __restrict__ src,
                           const int* __restrict__ dst, const float* __restrict__ dinv,
                           float* __restrict__ agg, int E) {
    int e = (int)((blockIdx.x * (size_t)blockDim.x + threadIdx.x) >> 5);
    int c = threadIdx.x & 31;
    if (e >= E) return;
    int s = src[e], d = dst[e];
    float nrm = dinv[s] * dinv[d];
    float v = hW[(size_t)s * HDIM + c] * nrm;
    atomicAdd(agg + (size_t)d * HDIM + c, v);
}

__global__ void k_relu(float* __restrict__ h, int n) {
    int i = blockIdx.x * blockDim.x + threadIdx.x;
    if (i < n) h[i] = fmaxf(h[i], 0.0f);
}

// ---------- scores: per-node dot with Wh, plus cash slot ----------
__global__ void k_score(const float* __restrict__ h2, const float* __restrict__ Wh,
                        const float* __restrict__ bh, const float* __restrict__ cash,
                        float* __restrict__ scores, int N) {
    int i = blockIdx.x * blockDim.x + threadIdx.x;
    if (i == 0) scores[0] = cash[0];
    if (i >= N) return;
    float s = bh[0];
    #pragma unroll
    for (int c = 0; c < HDIM; ++c) s += h2[(size_t)i * HDIM + c] * Wh[c];
    scores[i + 1] = s;
}

// ---------- softmax reductions ----------
__global__ void k_max(const float* __restrict__ s, int n, unsigned* __restrict__ maxbits) {
    __shared__ float sm[256];
    float m = -INFINITY;
    for (int i = blockIdx.x * blockDim.x + threadIdx.x; i < n; i += gridDim.x * blockDim.x)
        m = fmaxf(m, s[i]);
    sm[threadIdx.x] = m; __syncthreads();
    for (int o = 128; o > 0; o >>= 1) {
        if (threadIdx.x < o) sm[threadIdx.x] = fmaxf(sm[threadIdx.x], sm[threadIdx.x + o]);
        __syncthreads();
    }
    if (threadIdx.x == 0) atomicMax(maxbits, fflip(sm[0]));
}

__global__ void k_sumexp(const float* __restrict__ s, int n, const unsigned* __restrict__ maxbits,
                         float* __restrict__ sumf) {
    __shared__ float sm[256];
    float mx = funflip(*maxbits);
    float acc = 0.0f;
    for (int i = blockIdx.x * blockDim.x + threadIdx.x; i < n; i += gridDim.x * blockDim.x)
        acc += expf(s[i] - mx);
    sm[threadIdx.x] = acc; __syncthreads();
    for (int o = 128; o > 0; o >>= 1) {
        if (threadIdx.x < o) sm[threadIdx.x] += sm[threadIdx.x + o];
        __syncthreads();
    }
    if (threadIdx.x == 0) atomicAdd(sumf, sm[0]);
}

__global__ void k_finalize(const float* __restrict__ s, int n, const unsigned* __restrict__ maxbits,
                           const float* __restrict__ sumf, float* __restrict__ out) {
    int i = blockIdx.x * blockDim.x + threadIdx.x;
    if (i >= n) return;
    float mx = funflip(*maxbits);
    out[i] = expf(s[i] - mx) / (*sumf);
}

// ---------- launcher ----------
extern "C" void kernel_launch(void* const* d_in, const int* in_sizes, int n_in,
                              void* d_out, int out_size, void* d_ws, size_t ws_size,
                              hipStream_t stream) {
    const float* x    = (const float*)d_in[0];   // [N,128]
    const int*   ei   = (const int*)d_in[1];     // [2,E]
    const float* W1   = (const float*)d_in[2];   // [128,32]
    const float* b1   = (const float*)d_in[3];
    const float* W2   = (const float*)d_in[4];   // [32,32]
    const float* b2   = (const float*)d_in[5];
    const float* Wh   = (const float*)d_in[6];   // [32,1]
    const float* bh   = (const float*)d_in[7];
    const float* cash = (const float*)d_in[8];

    const int N  = in_sizes[0] / 128;
    const int E  = in_sizes[1] / 2;
    const int IN = 128;
    const int* src = ei;
    const int* dst = ei + E;

    // workspace carve-up (256B aligned)
    size_t off = 0;
    auto carve = [&](size_t bytes) { size_t o = off; off += (bytes + 255) & ~(size_t)255; return o; };
    unsigned char* w = (unsigned char*)d_ws;
    unsigned* deg     = (unsigned*)(w + carve((size_t)N * 4));
    float*    dinv    = (float*)   (w + carve((size_t)N * 4));
    float*    bufA    = (float*)   (w + carve((size_t)N * HDIM * 4));   // hW (layer transform)
    float*    bufB    = (float*)   (w + carve((size_t)N * HDIM * 4));   // agg / h
    float*    scores  = (float*)   (w + carve((size_t)(N + 1) * 4));
    unsigned* maxbits = (unsigned*)(w + carve(4));
    float*    sumf    = (float*)   (w + carve(4));
    (void)ws_size; (void)n_in; (void)out_size;

    const int T = 256;
    int nTiles     = (N + 15) / 16;                       // 16-row WMMA tiles
    int gemmBlocks = (nTiles + 3) / 4;                    // 4 waves (128 thr) per block
    int edgeBlocks = (E + 7) / 8;                         // 8 edges (waves) per 256-thr block

    // normalization
    k_init<<<(N + T - 1) / T, T, 0, stream>>>(deg, maxbits, sumf, N);
    k_deg <<<(E + T - 1) / T, T, 0, stream>>>(dst, deg, E);
    k_dinv<<<(N + T - 1) / T, T, 0, stream>>>(deg, dinv, N);

    // layer 1: hW = x @ W1 ; agg = self + edges ; relu
    k_gemm_wmma<<<gemmBlocks, 128, 0, stream>>>(x, W1, bufA, N, IN);
    k_agg_init <<<((size_t)N * HDIM + T - 1) / T, T, 0, stream>>>(bufA, dinv, b1, bufB, N);
    k_edge_agg <<<edgeBlocks, T, 0, stream>>>(bufA, src, dst, dinv, bufB, E);
    k_relu     <<<((size_t)N * HDIM + T - 1) / T, T, 0, stream>>>(bufB, N * HDIM);

    // layer 2: hW2 = h @ W2 ; agg ; relu
    k_gemm_wmma<<<gemmBlocks, 128, 0, stream>>>(bufB, W2, bufA, N, HDIM);
    k_agg_init <<<((size_t)N * HDIM + T - 1) / T, T, 0, stream>>>(bufA, dinv, b2, bufB, N);
    k_edge_agg <<<edgeBlocks, T, 0, stream>>>(bufA, src, dst, dinv, bufB, E);
    k_relu     <<<((size_t)N * HDIM + T - 1) / T, T, 0, stream>>>(bufB, N * HDIM);

    // scores + softmax
    k_score   <<<(N + T - 1) / T, T, 0, stream>>>(bufB, Wh, bh, cash, scores, N);
    k_max     <<<512, T, 0, stream>>>(scores, N + 1, maxbits);
    k_sumexp  <<<512, T, 0, stream>>>(scores, N + 1, maxbits, sumf);
    k_finalize<<<(N + 1 + T - 1) / T, T, 0, stream>>>(scores, N + 1, maxbits, sumf, (float*)d_out);
}